// RelationNetwork_2482491097868
// MI455X (gfx1250) — compile-verified
//
#include <hip/hip_runtime.h>
#include <hip/hip_bf16.h>
#include <math.h>

// Shapes (fixed by the reference): B=4, N=256, D=256, H=256; M = B*N = 1024.
#define Bn   4
#define Nn   256
#define Dn   256
#define Hn   256
#define Mn   (Bn * Nn)
#define LN_EPS 1e-5f

typedef __attribute__((ext_vector_type(2))) float v2f;
typedef __attribute__((ext_vector_type(8))) float v8f;

// ---------------------------------------------------------------------------
// Zero-fill helper (T accumulator must start at 0; harness poisons d_ws).
// ---------------------------------------------------------------------------
__global__ void rn_zero_f32(float* __restrict__ p, int n) {
    int i = blockIdx.x * blockDim.x + threadIdx.x;
    if (i < n) p[i] = 0.0f;
}

// ---------------------------------------------------------------------------
// FP32 GEMM via V_WMMA_F32_16X16X4_F32. One wave32 computes one 16x16 tile of
// C = X(1024x256) @ W(256x256), looping K in steps of 4 (64 WMMAs per tile).
// A-frag layout (16x4 f32): lane&15 = M row; lanes 0-15 hold K=k0+{0,1},
// lanes 16-31 hold K=k0+{2,3} (VGPR0=even K, VGPR1=odd K).
// B-frag layout (4x16 f32): lane&15 = N col; same K split across half-waves.
// C/D layout (16x16 f32, 8 VGPRs): row = m0 + r + (lane>=16 ? 8 : 0),
// col = h0 + (lane&15).
// ---------------------------------------------------------------------------
__global__ __launch_bounds__(256) void rn_gemm_f32_wmma(
    const float* __restrict__ X,   // (Mn, Dn) row-major
    const float* __restrict__ W,   // (Dn, Hn) row-major
    float* __restrict__ C)         // (Mn, Hn) row-major
{
    const int lane = threadIdx.x & 31;
    const int wave = threadIdx.x >> 5;
    const int tile = blockIdx.x * (blockDim.x >> 5) + wave; // 0..1023
    const int ht   = tile & (Hn / 16 - 1);   // 16 tiles along H
    const int mt   = tile / (Hn / 16);       // 64 tiles along M
    const int m0   = mt * 16;
    const int h0   = ht * 16;

    const int row  = m0 + (lane & 15);
    const int col  = h0 + (lane & 15);
    const int koff = (lane >> 4) * 2;        // 0 for lanes 0-15, 2 for 16-31

    const float* xrow = X + row * Dn;

    v8f c = {};
#pragma unroll 4
    for (int k0 = 0; k0 < Dn; k0 += 4) {
        v2f a, bfrag;
        a.x     = xrow[k0 + koff];
        a.y     = xrow[k0 + koff + 1];
        bfrag.x = W[(k0 + koff) * Hn + col];
        bfrag.y = W[(k0 + koff + 1) * Hn + col];
        // 8 args: (neg_a, A, neg_b, B, c_mod, C, reuse_a, reuse_b)
        c = __builtin_amdgcn_wmma_f32_16x16x4_f32(
            false, a, false, bfrag, (short)0, c, false, false);
    }

    const int mrow = m0 + ((lane >> 4) * 8);
#pragma unroll
    for (int r = 0; r < 8; ++r) {
        C[(mrow + r) * Hn + col] = c[r];
    }
}

// ---------------------------------------------------------------------------
// Fused pair stage. One wave32 per (b, i). Each lane owns 8 contiguous h.
// For every j: u = relu(a_i + b_j + bg); per-pair mean/var over H via wave
// shuffles; accumulate T[b,h] += rsqrt(var+eps) * (u - mean). Never touches
// the 67M-element pair tensor in memory. atomicAdd merges the 256 i-waves.
// ---------------------------------------------------------------------------
__global__ __launch_bounds__(128) void rn_pair_reduce(
    const float* __restrict__ A,    // (Bn*Nn, Hn)  i-part
    const float* __restrict__ Bm,   // (Bn*Nn, Hn)  j-part
    const float* __restrict__ bg,   // (Hn)
    float* __restrict__ T)          // (Bn, Hn) accumulator (pre-zeroed)
{
    const int lane  = threadIdx.x & 31;
    const int wid   = blockIdx.x * (blockDim.x >> 5) + (threadIdx.x >> 5); // 0..1023
    const int b     = wid >> 8;     // /256
    const int i     = wid & 255;
    const int hbase = lane * 8;     // 32B-aligned per lane -> b128 loads

    float af[8], acc[8];
    const float* arow = A + (b * Nn + i) * Hn + hbase;
#pragma unroll
    for (int r = 0; r < 8; ++r) {
        af[r]  = arow[r] + bg[hbase + r];
        acc[r] = 0.0f;
    }

    const float* bbase = Bm + b * Nn * Hn + hbase;
    for (int j = 0; j < Nn; ++j) {
        const float* brow = bbase + j * Hn;
        if (j + 1 < Nn) __builtin_prefetch(brow + Hn, 0, 3); // global_prefetch_b8

        float u[8];
        float s1 = 0.0f, s2 = 0.0f;
#pragma unroll
        for (int r = 0; r < 8; ++r) {
            float v = af[r] + brow[r];
            u[r] = v > 0.0f ? v : 0.0f;
            s1 += u[r];
            s2 += u[r] * u[r];
        }
        // wave32 all-reduce (sum) of s1, s2
#pragma unroll
        for (int off = 16; off >= 1; off >>= 1) {
            s1 += __shfl_xor(s1, off, 32);
            s2 += __shfl_xor(s2, off, 32);
        }
        const float m   = s1 * (1.0f / (float)Hn);
        const float var = s2 * (1.0f / (float)Hn) - m * m;
        const float sc  = rsqrtf(var + LN_EPS);
#pragma unroll
        for (int r = 0; r < 8; ++r) acc[r] += sc * (u[r] - m);
    }

#pragma unroll
    for (int r = 0; r < 8; ++r) {
        atomicAdd(&T[b * Hn + hbase + r], acc[r]);
    }
}

// ---------------------------------------------------------------------------
// Epilogue: r[b,h] = gamma*T + 65536*beta; rel = LN(relu(r@Wf + bf));
// out[b,n,h] = rel[b,h] + x[b,n,h]. One block (256 thr) per batch element.
// ---------------------------------------------------------------------------
__global__ __launch_bounds__(256) void rn_finalize(
    const float* __restrict__ T,
    const float* __restrict__ g_gamma, const float* __restrict__ g_beta,
    const float* __restrict__ Wf,      const float* __restrict__ bfv,
    const float* __restrict__ f_gamma, const float* __restrict__ f_beta,
    const float* __restrict__ x,
    float* __restrict__ out)
{
    __shared__ float rsh[Hn];
    __shared__ float red[Hn];

    const int h = threadIdx.x;
    const int b = blockIdx.x;

    // Undo the LN-sum factoring: r = gamma*T + (N*N)*beta
    const float rv = g_gamma[h] * T[b * Hn + h] +
                     (float)(Nn * Nn) * g_beta[h];
    rsh[h] = rv;
    __syncthreads();

    float y = bfv[h];
#pragma unroll 8
    for (int k = 0; k < Hn; ++k) {
        y += rsh[k] * Wf[k * Hn + h];
    }
    const float z = y > 0.0f ? y : 0.0f;

    // block-wide mean
    red[h] = z;
    __syncthreads();
    for (int s = Hn / 2; s >= 1; s >>= 1) {
        if (h < s) red[h] += red[h + s];
        __syncthreads();
    }
    const float m = red[0] * (1.0f / (float)Hn);
    __syncthreads();

    // block-wide second moment
    red[h] = z * z;
    __syncthreads();
    for (int s = Hn / 2; s >= 1; s >>= 1) {
        if (h < s) red[h] += red[h + s];
        __syncthreads();
    }
    const float var = red[0] * (1.0f / (float)Hn) - m * m;

    const float rel = (z - m) * rsqrtf(var + LN_EPS) * f_gamma[h] + f_beta[h];

    const float* xb = x + b * Nn * Hn;
    float* ob       = out + b * Nn * Hn;
    for (int n = 0; n < Nn; ++n) {
        ob[n * Hn + h] = rel + xb[n * Hn + h];
    }
}

// ---------------------------------------------------------------------------
extern "C" void kernel_launch(void* const* d_in, const int* in_sizes, int n_in,
                              void* d_out, int out_size, void* d_ws, size_t ws_size,
                              hipStream_t stream) {
    const float* x       = (const float*)d_in[0]; // (4,256,256)
    const float* Wg      = (const float*)d_in[1]; // (512,256)
    const float* bg      = (const float*)d_in[2]; // (256)
    const float* g_gamma = (const float*)d_in[3];
    const float* g_beta  = (const float*)d_in[4];
    const float* Wf      = (const float*)d_in[5]; // (256,256)
    const float* bfv     = (const float*)d_in[6];
    const float* f_gamma = (const float*)d_in[7];
    const float* f_beta  = (const float*)d_in[8];
    float* out           = (float*)d_out;         // (4,256,256)

    float* A_ws = (float*)d_ws;          // Mn*Hn = 262144 floats
    float* B_ws = A_ws + (size_t)Mn * Hn; // 262144 floats
    float* T_ws = B_ws + (size_t)Mn * Hn; // Bn*Hn = 1024 floats

    // 1) zero the pair accumulator
    rn_zero_f32<<<(Bn * Hn + 255) / 256, 256, 0, stream>>>(T_ws, Bn * Hn);

    // 2) the two fp32 WMMA GEMMs: a = x@Wg[:D], b = x@Wg[D:]
    //    1024 tiles, 8 waves/block -> 128 blocks each
    rn_gemm_f32_wmma<<<128, 256, 0, stream>>>(x, Wg, A_ws);
    rn_gemm_f32_wmma<<<128, 256, 0, stream>>>(x, Wg + (size_t)Dn * Hn, B_ws);

    // 3) fused relu+LayerNorm+sum over all 65536 (i,j) pairs per batch
    //    1024 waves total: 256 blocks x 4 waves
    rn_pair_reduce<<<256, 128, 0, stream>>>(A_ws, B_ws, bg, T_ws);

    // 4) epilogue GEMM + LN + broadcast add
    rn_finalize<<<Bn, Hn, 0, stream>>>(T_ws, g_gamma, g_beta, Wf, bfv,
                                       f_gamma, f_beta, x, out);
}